// Head_57509612094163
// MI455X (gfx1250) — compile-verified
//
#include <hip/hip_runtime.h>

typedef __attribute__((ext_vector_type(2))) float v2f;
typedef __attribute__((ext_vector_type(4))) float v4f;
typedef __attribute__((ext_vector_type(8))) float v8f;

#define NUM_MODES 64
#define NUM_KNOTS 8
#define NUM_STEPS 80
#define BATCH     8192
#define IN_STRIDE (NUM_MODES + NUM_MODES * NUM_KNOTS * 2) /* 1088 */

// ---------------------------------------------------------------------------
// Main spline kernel: one wave32 per 16-row tile of the flattened (b*64+m)
// dimension, 5 column tiles of 16 timesteps.
//
// pos(t) = t/16  =>  shift(t) = tile (uniform per 16-wide column tile),
//                    frac(t)  = (t&15)/16 (tile-invariant per lane).
// Each (dim, tile) is ONE exact V_WMMA_F32_16X16X4_F32:
//   A[M][K'] = knots[m][tile + K']  -> pair {s[tile], s[tile+1]} from a
//              12-cndmask sliding window s[j] = hi ? k[j+2] : k[j], j=0..5
//   B[K'][N] = basis_n[K']          -> tile-invariant, built once per lane
// Accumulator is seeded with the centroid tile; rotation + translation are
// applied on the D registers and stored as interleaved {px,py} pairs.
// ---------------------------------------------------------------------------
__global__ __launch_bounds__(256)
void spline_wmma_kernel(const float* __restrict__ inputs,
                        const float* __restrict__ xin,
                        const float* __restrict__ yin,
                        const float* __restrict__ yawin,
                        const float* __restrict__ centroids,
                        float* __restrict__ pred)
{
    const int lane   = threadIdx.x & 31;
    // wave-uniform by construction -> force scalar so address math rides SALU
    const int waveId = __builtin_amdgcn_readfirstlane(
                           (blockIdx.x * blockDim.x + threadIdx.x) >> 5); // 0..32767
    const int r0     = waveId << 4;         // first flattened row b*64+m
    const int b      = r0 >> 6;             // 64 modes per batch row
    const int mbase  = r0 & 63;             // multiple of 16 -> same b for tile

    const int  halfSel = lane >> 4;         // 0: lanes 0-15, 1: lanes 16-31
    const bool hi      = halfSel != 0;
    const int  nIdx    = lane & 15;

    // ---- preload all 8 knots per dim for row m = mbase + nIdx ----
    const float* rowPtr = inputs + (size_t)b * IN_STRIDE + NUM_MODES
                        + (size_t)(mbase + nIdx) * (2 * NUM_KNOTS);
    const v4f kxa = *(const v4f*)(rowPtr + 0);
    const v4f kxb = *(const v4f*)(rowPtr + 4);
    const v4f kya = *(const v4f*)(rowPtr + 8);
    const v4f kyb = *(const v4f*)(rowPtr + 12);

    // Sliding select window (named scalars only -> single cndmask each):
    // s[j] = hi ? k[j+2] : k[j].  Per-tile A operand = {s[tile], s[tile+1]}.
    const float sx0 = hi ? kxa.z : kxa.x;
    const float sx1 = hi ? kxa.w : kxa.y;
    const float sx2 = hi ? kxb.x : kxa.z;
    const float sx3 = hi ? kxb.y : kxa.w;
    const float sx4 = hi ? kxb.z : kxb.x;
    const float sx5 = hi ? kxb.w : kxb.y;
    const float sy0 = hi ? kya.z : kya.x;
    const float sy1 = hi ? kya.w : kya.y;
    const float sy2 = hi ? kyb.x : kya.z;
    const float sy3 = hi ? kyb.y : kya.w;
    const float sy4 = hi ? kyb.z : kyb.x;
    const float sy5 = hi ? kyb.w : kyb.y;
    const float sx[6] = { sx0, sx1, sx2, sx3, sx4, sx5 };   // constant-indexed only
    const float sy[6] = { sy0, sy1, sy2, sy3, sy4, sy5 };

    // ---- cubic B-spline basis at frac = nIdx/16 (tile-invariant) ----
    const float f   = (float)nIdx * 0.0625f;
    const float f2  = f * f;
    const float f3  = f2 * f;
    const float omf = 1.0f - f;
    const float bb0 = omf * omf * omf * (1.0f / 6.0f);
    const float bb1 = (3.0f * f3 - 6.0f * f2 + 4.0f) * (1.0f / 6.0f);
    const float bb2 = (-3.0f * f3 + 3.0f * f2 + 3.0f * f + 1.0f) * (1.0f / 6.0f);
    const float bb3 = f3 * (1.0f / 6.0f);

    // B operand (4x16): VGPR0 -> K'=2*halfSel, VGPR1 -> K'=2*halfSel+1
    v2f bw;
    bw.x = hi ? bb2 : bb0;
    bw.y = hi ? bb3 : bb1;

    // ---- wave-uniform rotation/translation ----
    const float yaw = yawin[b];
    const float rc  = __cosf(yaw);
    const float rs  = __sinf(yaw);
    const float nrs = -rs;
    const float tx  = xin[b];
    const float ty  = yin[b];

    const float* cbase = centroids + (size_t)(mbase + 8 * halfSel) * (2 * NUM_STEPS);
    float* outBase0 = pred + 2 * ((size_t)(r0 + 8 * halfSel) * NUM_STEPS);

    #pragma unroll
    for (int tile = 0; tile < 5; ++tile) {
        const int t = tile * 16 + nIdx;     // this lane's column (N)

        // A operands: register pairs from the sliding window (no per-tile VALU)
        v2f ax, ay;
        ax.x = sx[tile];  ax.y = sx[tile + 1];
        ay.x = sy[tile];  ay.y = sy[tile + 1];

        // seed accumulators with centroid tile (rows M = v + 8*halfSel)
        v8f cx, cyv;
        #pragma unroll
        for (int v = 0; v < 8; ++v) {
            const v2f cc = *(const v2f*)(cbase + (size_t)v * (2 * NUM_STEPS) + 2 * t);
            cx[v]  = cc.x;
            cyv[v] = cc.y;
        }

        // D = A*B + C : one exact 16x16x4 f32 WMMA per dim
        const v8f dx = __builtin_amdgcn_wmma_f32_16x16x4_f32(false, ax, false, bw,
                                                             (short)0, cx, false, false);
        const v8f dy = __builtin_amdgcn_wmma_f32_16x16x4_f32(false, ay, false, bw,
                                                             (short)0, cyv, false, false);

        // rotate + translate, store interleaved {px, py} pairs
        #pragma unroll
        for (int v = 0; v < 8; ++v) {
            const float fx = dx[v];
            const float fy = dy[v];
            v2f o;
            o.x = fmaf(rc, fx, fmaf(nrs, fy, tx));
            o.y = fmaf(rs, fx, fmaf(rc,  fy, ty));
            *(v2f*)(outBase0 + (size_t)v * (2 * NUM_STEPS) + 2 * t) = o;
        }
    }
}

// ---------------------------------------------------------------------------
// Softmax over 64 logits per batch row: one wave32 per row, 2 values/lane.
// ---------------------------------------------------------------------------
__global__ __launch_bounds__(256)
void softmax_kernel(const float* __restrict__ inputs, float* __restrict__ conf)
{
    const int lane = threadIdx.x & 31;
    const int row  = (blockIdx.x * blockDim.x + threadIdx.x) >> 5;
    if (row >= BATCH) return;
    const float* p = inputs + (size_t)row * IN_STRIDE;
    const float a = p[lane];
    const float c = p[lane + 32];
    float mx = fmaxf(a, c);
    #pragma unroll
    for (int off = 16; off > 0; off >>= 1)
        mx = fmaxf(mx, __shfl_xor(mx, off, 32));
    const float ea = __expf(a - mx);
    const float ec = __expf(c - mx);
    float s = ea + ec;
    #pragma unroll
    for (int off = 16; off > 0; off >>= 1)
        s += __shfl_xor(s, off, 32);
    const float inv = 1.0f / s;
    conf[(size_t)row * 64 + lane]      = ea * inv;
    conf[(size_t)row * 64 + lane + 32] = ec * inv;
}

// ---------------------------------------------------------------------------
// knots output = inputs[:, 64:] row-major copy (reshape is a no-op). float4.
// ---------------------------------------------------------------------------
__global__ __launch_bounds__(256)
void knots_copy_kernel(const float* __restrict__ inputs, float* __restrict__ knotsOut)
{
    const int tid = blockIdx.x * blockDim.x + threadIdx.x;  // B*256 threads
    const int b   = tid >> 8;
    const int j4  = tid & 255;
    const float4* src = (const float4*)(inputs + (size_t)b * IN_STRIDE + NUM_MODES) + j4;
    float4*       dst = (float4*)(knotsOut + (size_t)b * 1024) + j4;
    *dst = *src;
}

extern "C" void kernel_launch(void* const* d_in, const int* in_sizes, int n_in,
                              void* d_out, int out_size, void* d_ws, size_t ws_size,
                              hipStream_t stream)
{
    const float* inputs    = (const float*)d_in[0];
    const float* x         = (const float*)d_in[1];
    const float* y         = (const float*)d_in[2];
    const float* yaw       = (const float*)d_in[3];
    const float* centroids = (const float*)d_in[4];

    float* out   = (float*)d_out;
    float* pred  = out;                                                  // B*64*80*2
    float* conf  = out + (size_t)BATCH * NUM_MODES * NUM_STEPS * 2;      // B*64
    float* knots = conf + (size_t)BATCH * NUM_MODES;                     // B*64*2*8

    // 32768 row-tiles, 8 waves per 256-thread block -> 4096 blocks (exact fill,
    // EXEC all-ones for every WMMA wave).
    spline_wmma_kernel<<<4096, 256, 0, stream>>>(inputs, x, y, yaw, centroids, pred);
    // 8192 rows, 8 waves/block -> 1024 blocks
    softmax_kernel<<<1024, 256, 0, stream>>>(inputs, conf);
    // 8192 * 256 float4 copies
    knots_copy_kernel<<<8192, 256, 0, stream>>>(inputs, knots);
}